// NonLocalInteraction_49160195670228
// MI455X (gfx1250) — compile-verified
//
#include <hip/hip_runtime.h>

#define D      128
#define MFEAT  128
#define G      64
#define NTOT   8192
#define LDSW   132            // 128 + 4 pad (bank-conflict stagger on 64 banks)
#define EPSK   1e-4f
#define DNORM  0.29730177875068026f   // 128^-0.25
#define RATIO  0.08838834764831843f   // 128^-0.5

typedef __attribute__((ext_vector_type(2))) float v2f;
typedef __attribute__((ext_vector_type(8))) float v8f;

__device__ __forceinline__ float swish1(float x, float a, float b) {
  return a * x * (1.0f / (1.0f + __expf(-b * x)));
}

__device__ __forceinline__ void zero_acc(v8f acc[8]) {
#pragma unroll
  for (int t = 0; t < 8; ++t)
#pragma unroll
    for (int r = 0; r < 8; ++r) acc[t][r] = 0.0f;
}

// One wave computes a 16-row x 128-col stripe: out[m0..m0+16, 0..128) +=
// A[16 x 128] * B[128 x 128], K accumulated with V_WMMA_F32_16X16X4_F32.
// ATRANS: A element (m,k) read as A[k*lda + m] (for kp^T).
// BROWK : B element (k,n) read as B[k*ldb + n]  (row-major over k),
//         else B(k,n) = B[n*ldb + k]            (i.e. X @ W.T pattern).
template <bool ATRANS, bool BROWK>
__device__ __forceinline__ void gemm_stripe(const float* __restrict__ A, int lda,
                                            const float* __restrict__ B, int ldb,
                                            int m0, int lane, v8f acc[8]) {
  const int mr = lane & 15;          // matrix row (A) / col (B) within tile
  const int kb = (lane >> 4) << 1;   // k-pair base for this half-wave
#pragma unroll 4
  for (int k0 = 0; k0 < D; k0 += 4) {
    const int k = k0 + kb;
    v2f a;
    if (ATRANS) {
      a.x = A[k * lda + m0 + mr];
      a.y = A[(k + 1) * lda + m0 + mr];
    } else {
      a.x = A[(m0 + mr) * lda + k];
      a.y = A[(m0 + mr) * lda + k + 1];
    }
#pragma unroll
    for (int t = 0; t < 8; ++t) {
      const int col = t * 16 + mr;
      v2f b;
      if (BROWK) {
        b.x = B[k * ldb + col];
        b.y = B[(k + 1) * ldb + col];
      } else {
        b.x = B[col * ldb + k];
        b.y = B[col * ldb + k + 1];
      }
      acc[t] = __builtin_amdgcn_wmma_f32_16x16x4_f32(false, a, false, b,
                                                     (short)0, acc[t],
                                                     false, false);
    }
  }
}

// ---------------- Kernel 1: Q/K/V via three ResidualMLPs ----------------
// grid = (N/128, 3); block = 256 (8 wave32). LDS: two 128x132 f32 buffers.
__global__ __launch_bounds__(256)
void qkv_kernel(const float* __restrict__ x,
                const float* __restrict__ W1,
                const float* __restrict__ W2,
                const float* __restrict__ W3,
                const float* __restrict__ alpha,
                const float* __restrict__ beta,
                float* __restrict__ qkv /* [3][N][D] */) {
  __shared__ float buf0[128 * LDSW];
  __shared__ float buf1[128 * LDSW];

  const int s = blockIdx.y;
  const int rowBase = blockIdx.x * 128;
  const int tid = threadIdx.x;
  const int lane = tid & 31;
  const int wave = tid >> 5;
  const int m0 = wave * 16;

  const float* W1s = W1 + s * D * D;
  const float* W2s = W2 + s * D * D;
  const float* W3s = W3 + s * D * D;
  const float* a0 = alpha + (s * 3 + 0) * D;
  const float* a1 = alpha + (s * 3 + 1) * D;
  const float* a2 = alpha + (s * 3 + 2) * D;
  const float* b0 = beta + (s * 3 + 0) * D;
  const float* b1 = beta + (s * 3 + 1) * D;
  const float* b2 = beta + (s * 3 + 2) * D;

  // stage: buf0 = swish(x, a0, b0)
  for (int i = tid; i < 128 * D; i += 256) {
    const int r = i >> 7, c = i & 127;
    buf0[r * LDSW + c] = swish1(x[(rowBase + r) * D + c], a0[c], b0[c]);
  }
  __syncthreads();

  v8f acc[8];
  const int half = (lane >> 4) * 8;
  const int cn = lane & 15;

  // GEMM1: buf1 = buf0 @ W1.T
  zero_acc(acc);
  gemm_stripe<false, false>(buf0, LDSW, W1s, D, m0, lane, acc);
#pragma unroll
  for (int t = 0; t < 8; ++t)
#pragma unroll
    for (int r = 0; r < 8; ++r)
      buf1[(m0 + half + r) * LDSW + t * 16 + cn] = acc[t][r];
  __syncthreads();

  // buf0 = swish(buf1, a1, b1)
  for (int i = tid; i < 128 * D; i += 256) {
    const int r = i >> 7, c = i & 127;
    buf0[r * LDSW + c] = swish1(buf1[r * LDSW + c], a1[c], b1[c]);
  }
  __syncthreads();

  // GEMM2: buf1 = buf0 @ W2.T
  zero_acc(acc);
  gemm_stripe<false, false>(buf0, LDSW, W2s, D, m0, lane, acc);
#pragma unroll
  for (int t = 0; t < 8; ++t)
#pragma unroll
    for (int r = 0; r < 8; ++r)
      buf1[(m0 + half + r) * LDSW + t * 16 + cn] = acc[t][r];
  __syncthreads();

  // residual + swish: buf0 = swish(x + buf1, a2, b2)
  for (int i = tid; i < 128 * D; i += 256) {
    const int r = i >> 7, c = i & 127;
    const float v = x[(rowBase + r) * D + c] + buf1[r * LDSW + c];
    buf0[r * LDSW + c] = swish1(v, a2[c], b2[c]);
  }
  __syncthreads();

  // GEMM3: qkv[s] = buf0 @ W3.T  (store straight to workspace)
  zero_acc(acc);
  gemm_stripe<false, false>(buf0, LDSW, W3s, D, m0, lane, acc);
  float* outp = qkv + (size_t)s * NTOT * D;
#pragma unroll
  for (int t = 0; t < 8; ++t)
#pragma unroll
    for (int r = 0; r < 8; ++r)
      outp[(size_t)(rowBase + m0 + half + r) * D + t * 16 + cn] = acc[t][r];
}

// ---------------- Kernel 2: fused Performer attention, one WG per graph ----
__global__ __launch_bounds__(256)
void attn_kernel(const float* __restrict__ Q,
                 const float* __restrict__ K,
                 const float* __restrict__ V,
                 const float* __restrict__ proj,
                 const int* __restrict__ num_atoms,
                 float* __restrict__ out) {
  __shared__ float qb[128 * LDSW];   // Q*dnorm -> qp
  __shared__ float kb[128 * LDSW];   // K*dnorm -> kp -> kv
  __shared__ float qdiag[128], kdiag[128], ksum[128], dinv[128], rowArr[128];
  __shared__ int gstartSh, RSh;
  __shared__ float kmaxSh;

  const int g = blockIdx.x;
  const int tid = threadIdx.x;
  const int lane = tid & 31;
  const int wave = tid >> 5;
  const int m0 = wave * 16;
  const int half = (lane >> 4) * 8;
  const int cn = lane & 15;

  if (tid == 0) {
    int st = 0;
    for (int i = 0; i < g; ++i) st += num_atoms[i];
    gstartSh = st;
    RSh = num_atoms[g];
  }
  __syncthreads();
  const int gstart = gstartSh;
  const int R = RSh;

  // stage scaled Q/K (rows >= R zeroed)
  for (int i = tid; i < 128 * D; i += 256) {
    const int r = i >> 7, c = i & 127;
    float qv = 0.0f, kvv = 0.0f;
    if (r < R) {
      qv = Q[(size_t)(gstart + r) * D + c] * DNORM;
      kvv = K[(size_t)(gstart + r) * D + c] * DNORM;
    }
    qb[r * LDSW + c] = qv;
    kb[r * LDSW + c] = kvv;
  }
  __syncthreads();

  if (tid < 128) {
    float sq = 0.0f, sk = 0.0f;
    for (int d = 0; d < D; ++d) {
      const float a = qb[tid * LDSW + d];
      const float b = kb[tid * LDSW + d];
      sq += a * a;
      sk += b * b;
    }
    qdiag[tid] = 0.5f * sq;   // 0.5*sum(Q^2)*dnorm^2 (qb is pre-scaled)
    kdiag[tid] = 0.5f * sk;
  }
  __syncthreads();

  v8f acc[8];
  float rm[8];

  // ---- kd = kb @ proj.T ; per-graph max ; kp in-place into kb ----
  zero_acc(acc);
  gemm_stripe<false, false>(kb, LDSW, proj, D, m0, lane, acc);
#pragma unroll
  for (int r = 0; r < 8; ++r) {
    float mx = acc[0][r];
#pragma unroll
    for (int t = 1; t < 8; ++t) mx = fmaxf(mx, acc[t][r]);
    rm[r] = mx;
  }
#pragma unroll
  for (int r = 0; r < 8; ++r)
    for (int off = 1; off < 16; off <<= 1)
      rm[r] = fmaxf(rm[r], __shfl_xor(rm[r], off, 32));
  if ((lane & 15) == 0) {
#pragma unroll
    for (int r = 0; r < 8; ++r) rowArr[m0 + half + r] = rm[r];
  }
  __syncthreads();
  if (tid == 0) {
    float mx = -3.402823466e38f;
    for (int i = 0; i < R; ++i) mx = fmaxf(mx, rowArr[i]);
    kmaxSh = mx;
  }
  __syncthreads();
  {
    const float km = kmaxSh;
#pragma unroll
    for (int t = 0; t < 8; ++t)
#pragma unroll
      for (int r = 0; r < 8; ++r) {
        const int row = m0 + half + r;
        float v = 0.0f;
        if (row < R)
          v = RATIO * (__expf(acc[t][r] - kdiag[row] - km) + EPSK);
        kb[row * LDSW + t * 16 + cn] = v;   // this wave's own rows only
      }
  }

  // ---- qd = qb @ proj.T ; per-row max ; qp in-place into qb ----
  zero_acc(acc);
  gemm_stripe<false, false>(qb, LDSW, proj, D, m0, lane, acc);
#pragma unroll
  for (int r = 0; r < 8; ++r) {
    float mx = acc[0][r];
#pragma unroll
    for (int t = 1; t < 8; ++t) mx = fmaxf(mx, acc[t][r]);
    rm[r] = mx;
  }
#pragma unroll
  for (int r = 0; r < 8; ++r)
    for (int off = 1; off < 16; off <<= 1)
      rm[r] = fmaxf(rm[r], __shfl_xor(rm[r], off, 32));
#pragma unroll
  for (int t = 0; t < 8; ++t)
#pragma unroll
    for (int r = 0; r < 8; ++r) {
      const int row = m0 + half + r;
      float v = 0.0f;
      if (row < R)
        v = RATIO * (__expf(acc[t][r] - qdiag[row] - rm[r]) + EPSK);
      qb[row * LDSW + t * 16 + cn] = v;
    }
  __syncthreads();   // qp & kp fully materialized

  // ksum[m] = sum_n kp[n,m]
  if (tid < 128) {
    float sv = 0.0f;
    for (int n = 0; n < 128; ++n) sv += kb[n * LDSW + tid];
    ksum[tid] = sv;
  }
  __syncthreads();

  // kv = kp^T @ V_g  (accumulators held in regs across barrier, then
  // written back over kb — kp is dead after ksum)
  zero_acc(acc);
  gemm_stripe<true, true>(kb, LDSW, V + (size_t)gstart * D, D, m0, lane, acc);
  __syncthreads();
#pragma unroll
  for (int t = 0; t < 8; ++t)
#pragma unroll
    for (int r = 0; r < 8; ++r)
      kb[(m0 + half + r) * LDSW + t * 16 + cn] = acc[t][r];
  __syncthreads();

  // dinv[n] = 1 / (qp[n,:] . ksum)
  if (tid < 128) {
    float sv = 0.0f;
    for (int m = 0; m < MFEAT; ++m) sv += qb[tid * LDSW + m] * ksum[m];
    dinv[tid] = 1.0f / sv;
  }
  __syncthreads();

  // out = (qp @ kv) * dinv
  zero_acc(acc);
  gemm_stripe<false, true>(qb, LDSW, kb, LDSW, m0, lane, acc);
#pragma unroll
  for (int t = 0; t < 8; ++t)
#pragma unroll
    for (int r = 0; r < 8; ++r) {
      const int row = m0 + half + r;
      if (row < R)
        out[(size_t)(gstart + row) * D + t * 16 + cn] = acc[t][r] * dinv[row];
    }
}

extern "C" void kernel_launch(void* const* d_in, const int* in_sizes, int n_in,
                              void* d_out, int out_size, void* d_ws, size_t ws_size,
                              hipStream_t stream) {
  (void)in_sizes; (void)n_in; (void)out_size; (void)ws_size;
  const float* x         = (const float*)d_in[0];
  const int*   num_atoms = (const int*)d_in[1];
  const float* W1        = (const float*)d_in[2];
  const float* W2        = (const float*)d_in[3];
  const float* W3        = (const float*)d_in[4];
  const float* alpha     = (const float*)d_in[5];
  const float* beta      = (const float*)d_in[6];
  const float* proj      = (const float*)d_in[7];
  float* out = (float*)d_out;

  float* ws = (float*)d_ws;          // [3][N][D] fp32 = 12.6 MB scratch
  float* Q = ws;
  float* K = ws + (size_t)NTOT * D;
  float* V = ws + 2 * (size_t)NTOT * D;

  dim3 grid1(NTOT / 128, 3);
  qkv_kernel<<<grid1, 256, 0, stream>>>(x, W1, W2, W3, alpha, beta, ws);
  attn_kernel<<<G, 256, 0, stream>>>(Q, K, V, proj, num_atoms, out);
}